// LoadBalancingLoss_59141699666465
// MI455X (gfx1250) — compile-verified
//
#include <hip/hip_runtime.h>
#include <hip/hip_bf16.h>

#define NUM_EXPERTS 64
#define NSUB 8          // LDS sub-histograms per block to reduce ds-atomic contention

typedef __attribute__((ext_vector_type(2))) float v2f;
typedef __attribute__((ext_vector_type(8))) float v8f;

// ---------------------------------------------------------------------------
// Workspace layout (floats): ws[0..63] = expert counts, ws[64..127] = gate sums
// ---------------------------------------------------------------------------

__global__ void lbl_zero_ws(float* __restrict__ ws) {
  if (threadIdx.x < 2 * NUM_EXPERTS) ws[threadIdx.x] = 0.0f;
}

// Bandwidth-bound histogram pass: b128 loads, ds_add_f32 LDS atomics,
// global_atomic_add_f32 reduction, global_prefetch_b8 one stride ahead.
__global__ __launch_bounds__(256) void lbl_hist(const float* __restrict__ gates,
                                                const int*   __restrict__ idx,
                                                float*       __restrict__ ws,
                                                long long n) {
  __shared__ float s_cnt[NSUB * NUM_EXPERTS];
  __shared__ float s_sum[NSUB * NUM_EXPERTS];
  for (int i = threadIdx.x; i < NSUB * NUM_EXPERTS; i += blockDim.x) {
    s_cnt[i] = 0.0f;
    s_sum[i] = 0.0f;
  }
  __syncthreads();

  const int sub = (threadIdx.x & (NSUB - 1)) * NUM_EXPERTS;
  const long long stride = (long long)gridDim.x * blockDim.x * 4;
  long long i = ((long long)blockIdx.x * blockDim.x + threadIdx.x) * 4;

  for (; i + 3 < n; i += stride) {
    // gfx1250 speculative prefetch for the next grid-stride iteration
    __builtin_prefetch(gates + i + stride, 0, 3);
    __builtin_prefetch(idx + i + stride, 0, 3);

    const float4 g = *reinterpret_cast<const float4*>(gates + i);
    const int4   e = *reinterpret_cast<const int4*>(idx + i);

    atomicAdd(&s_cnt[sub + (e.x & 63)], 1.0f);
    atomicAdd(&s_sum[sub + (e.x & 63)], g.x);
    atomicAdd(&s_cnt[sub + (e.y & 63)], 1.0f);
    atomicAdd(&s_sum[sub + (e.y & 63)], g.y);
    atomicAdd(&s_cnt[sub + (e.z & 63)], 1.0f);
    atomicAdd(&s_sum[sub + (e.z & 63)], g.z);
    atomicAdd(&s_cnt[sub + (e.w & 63)], 1.0f);
    atomicAdd(&s_sum[sub + (e.w & 63)], g.w);
  }
  // tail (only the unique thread whose 4-chunk straddles n does work)
  if (i < n) {
    for (long long j = i; j < n && j < i + 4; ++j) {
      atomicAdd(&s_cnt[sub + (idx[j] & 63)], 1.0f);
      atomicAdd(&s_sum[sub + (idx[j] & 63)], gates[j]);
    }
  }
  __syncthreads();

  for (int e = threadIdx.x; e < NUM_EXPERTS; e += blockDim.x) {
    float c = 0.0f, s = 0.0f;
#pragma unroll
    for (int j = 0; j < NSUB; ++j) {
      c += s_cnt[j * NUM_EXPERTS + e];
      s += s_sum[j * NUM_EXPERTS + e];
    }
    unsafeAtomicAdd(&ws[e], c);                 // global_atomic_add_f32
    unsafeAtomicAdd(&ws[NUM_EXPERTS + e], s);
  }
}

// Finalize with the matrix unit: A(16x4) = counts/N, B(4x16) = gateSums/N laid
// out transposed so D[m][n] = sum_k cnt[m*4+k]*gs[n*4+k]; trace(D) is the
// length-64 dot product. Single wave32, EXEC all ones (no divergence pre-WMMA).
__global__ void lbl_finalize(const float* __restrict__ ws,
                             float* __restrict__ out,
                             float inv_total) {
  const int lane = threadIdx.x;          // 0..31, one wave
  const int m    = lane & 15;            // A row / B column
  const int k0   = (lane < 16) ? 0 : 2;  // lane-half selects K pair (ISA 7.12.2)

  v2f a, b;
  a.x = ws[m * 4 + k0]               * inv_total;   // A[m][k0]   = cnt[m*4+k0]/N
  a.y = ws[m * 4 + k0 + 1]           * inv_total;   // A[m][k0+1]
  b.x = ws[NUM_EXPERTS + m * 4 + k0]     * inv_total; // B[k0][m]   = gs[m*4+k0]/N
  b.y = ws[NUM_EXPERTS + m * 4 + k0 + 1] * inv_total; // B[k0+1][m]

  v8f c = {};
  c = __builtin_amdgcn_wmma_f32_16x16x4_f32(
      /*neg_a=*/false, a, /*neg_b=*/false, b,
      /*c_mod=*/(short)0, c, /*reuse_a=*/false, /*reuse_b=*/false);

  // Diagonal of D: m<8 -> lanes 0..7 VGPR m ; m>=8 -> lanes 24..31 VGPR m-8
  float diag = 0.0f;
  if (lane < 8)        diag = c[lane];
  else if (lane >= 24) diag = c[lane - 24];

  for (int off = 16; off > 0; off >>= 1)
    diag += __shfl_xor(diag, off, 32);

  if (lane == 0) out[0] = (float)NUM_EXPERTS * diag;
}

extern "C" void kernel_launch(void* const* d_in, const int* in_sizes, int n_in,
                              void* d_out, int out_size, void* d_ws, size_t ws_size,
                              hipStream_t stream) {
  const float* gates   = (const float*)d_in[0];
  const int*   indices = (const int*)d_in[1];
  float*       out     = (float*)d_out;
  float*       ws      = (float*)d_ws;

  const long long n = (long long)in_sizes[0];   // B*S*K = total token-expert pairs
  const float inv_total = 1.0f / (float)n;

  lbl_zero_ws<<<1, 128, 0, stream>>>(ws);

  const int threads = 256;
  int blocks = (int)((n / 4 + threads - 1) / threads);
  if (blocks > 1024) blocks = 1024;   // grid-stride; ~8K wave32s fills the chip
  if (blocks < 1)    blocks = 1;
  lbl_hist<<<blocks, threads, 0, stream>>>(gates, indices, ws, n);

  lbl_finalize<<<1, 32, 0, stream>>>(ws, out, inv_total);
}